// TAGLayer_39788577030286
// MI455X (gfx1250) — compile-verified
//
#include <hip/hip_runtime.h>
#include <math.h>

typedef __attribute__((ext_vector_type(2))) float v2f;
typedef __attribute__((ext_vector_type(8))) float v8f;

// x layout: (N=16, C=64, T=256, V=25, M=12), strides:
//   j:1  v:12  t:300  c:76800  n:4915200
#define S_V 12
#define S_T 300
#define S_C 76800
#define S_N 4915200

#define BIGV 1000000.0f
#define TAU  0.35f

// Row index r in [0,1600) maps to (c = r/25, v = r%25); per wave-iteration r
// advances by 128 = 5*25 + 3:
#define R_STEP_OFF (5 * S_C + 3 * S_V)   // 384036 floats
#define V_WRAP_OFF (S_C - 25 * S_V)      // 76500 floats (extra when v wraps)

// ---------------------------------------------------------------------------
// Kernel 1: one wave (32 threads) per (n,t); builds normalized 12x12 adjacency
// and stores the fused operator  B[i][j] = delta_ij + lam * A[i][j]
// zero-padded to 16x16 row-major into ws[nt*256].  Kernel 2's WMMA result
// X * B^T then equals x + lam*y directly (no residual pass).
// ---------------------------------------------------------------------------
__global__ __launch_bounds__(32)
void tag_build_adj(const float* __restrict__ x, const float* __restrict__ lamp,
                   float* __restrict__ Aout)
{
    const int nt   = blockIdx.x;          // 0..4095
    const int n    = nt >> 8;
    const int t    = nt & 255;
    const int lane = threadIdx.x;         // 0..31
    const int base = n * S_N + t * S_T;

    __shared__ float ctr[12][3];
    __shared__ float pres[12];
    __shared__ float dstm[12][12];
    __shared__ float adjW[12][12];
    __shared__ float Bm[12][12];
    __shared__ float rs[12];
    __shared__ float dis[12];

    // centers (mean over V of channels 0..2) + presence (sum of |.| over d,v)
    if (lane < 12) {
        float s0 = 0.f, s1 = 0.f, s2 = 0.f, mag = 0.f;
        #pragma unroll 5
        for (int v = 0; v < 25; ++v) {
            float c0 = x[base + 0 * S_C + v * S_V + lane];
            float c1 = x[base + 1 * S_C + v * S_V + lane];
            float c2 = x[base + 2 * S_C + v * S_V + lane];
            s0 += c0; s1 += c1; s2 += c2;
            mag += fabsf(c0) + fabsf(c1) + fabsf(c2);
        }
        ctr[lane][0] = s0 * (1.0f / 25.0f);
        ctr[lane][1] = s1 * (1.0f / 25.0f);
        ctr[lane][2] = s2 * (1.0f / 25.0f);
        pres[lane]   = (mag > 1e-6f) ? 1.0f : 0.0f;
    }
    __syncthreads();

    // pairwise distances + masking; zero adjW
    for (int p = lane; p < 144; p += 32) {
        const int i = p / 12, j = p - 12 * (p / 12);
        float dx = ctr[i][0] - ctr[j][0];
        float dy = ctr[i][1] - ctr[j][1];
        float dz = ctr[i][2] - ctr[j][2];
        float d  = sqrtf(fmaxf(dx * dx + dy * dy + dz * dz, 0.0f));
        if (!(pres[i] > 0.0f && pres[j] > 0.0f)) d = BIGV;
        if (i == j) d += BIGV;
        dstm[i][j] = d;
        adjW[i][j] = 0.0f;
    }
    __syncthreads();

    // per-row top-2 (first-occurrence min matches top_k tie order) + weights
    if (lane < 12) {
        int i1 = 0; float d1 = 3.0e38f;
        #pragma unroll
        for (int j = 0; j < 12; ++j) {
            float d = dstm[lane][j];
            if (d < d1) { d1 = d; i1 = j; }
        }
        int i2 = (i1 == 0) ? 1 : 0; float d2 = 3.0e38f;
        #pragma unroll
        for (int j = 0; j < 12; ++j) {
            if (j == i1) continue;
            float d = dstm[lane][j];
            if (d < d2) { d2 = d; i2 = j; }
        }
        float w1 = expf(-d1 / TAU);
        float w2 = expf(-d2 / TAU);
        float s  = w1 + w2 + 1e-6f;
        adjW[lane][i1] = w1 / s;
        adjW[lane][i2] = w2 / s;
    }
    __syncthreads();

    // symmetrize + self loop
    for (int p = lane; p < 144; p += 32) {
        const int i = p / 12, j = p - 12 * (p / 12);
        float b = 0.5f * (adjW[i][j] + adjW[j][i]);
        if (i == j) b += 0.5f;
        Bm[i][j] = b;
    }
    __syncthreads();

    // row sums for row-normalization
    if (lane < 12) {
        float s = 0.f;
        #pragma unroll
        for (int j = 0; j < 12; ++j) s += Bm[lane][j];
        rs[lane] = 1.0f / (s + 1e-6f);
    }
    __syncthreads();

    // row normalize in place, compute deg^{-1/2}
    if (lane < 12) {
        const float r = rs[lane];
        float s2 = 0.f;
        #pragma unroll
        for (int j = 0; j < 12; ++j) {
            float v = Bm[lane][j] * r;
            Bm[lane][j] = v;
            s2 += v;
        }
        dis[lane] = 1.0f / sqrtf(fmaxf(s2, 1e-6f));
    }
    __syncthreads();

    // store padded 16x16 fused operator: delta_ij + lam * dis_i * C_ij * dis_j
    const float lam = lamp[0];
    float* outp = Aout + nt * 256;
    for (int p = lane; p < 256; p += 32) {
        const int i = p >> 4, j = p & 15;
        float v = 0.0f;
        if (i < 12 && j < 12) {
            v = lam * (dis[i] * Bm[i][j] * dis[j]);
            if (i == j) v += 1.0f;
        }
        outp[p] = v;
    }
}

// ---------------------------------------------------------------------------
// Kernel 2: out = X_tile @ (I + lam*A)^T via V_WMMA_F32_16X16X4_F32.
// Block = 256 threads (8 waves) per (n,t); each wave handles every 8th
// 16-row tile of the 1600x12 per-(n,t) matrix (row r = c*25 + v).
// All addressing is 32-bit element indices (SADDR + scaled voffset form);
// the per-g store offsets are independent selects (<= 1 wrap per 8 rows),
// not a dependence chain.
// ---------------------------------------------------------------------------
__global__ __launch_bounds__(256)
void tag_apply_wmma(const float* __restrict__ x, const float* __restrict__ Aw,
                    float* __restrict__ out)
{
    const int nt   = blockIdx.x;
    const int n    = nt >> 8;
    const int t    = nt & 255;
    const int tid  = threadIdx.x;
    const int lane = tid & 31;
    const int wave = tid >> 5;
    const int base = n * S_N + t * S_T;

    const int icol  = lane & 15;        // A-op row M, and D column N
    const int khalf = (lane >> 4) << 1; // 0 for lanes 0-15, 2 for lanes 16-31
    const int mbase = (lane >> 4) * 8;  // first D row held by this lane half

    // B operand: Bop[k][i] = ws[i][k] (ws row i = 16 contiguous floats).
    const v2f* bp = (const v2f*)(Aw + nt * 256 + icol * 16 + khalf);
    const v2f b0 = bp[0];  // k = 0+khalf, 1+khalf
    const v2f b1 = bp[2];  // k = 4+khalf, 5+khalf
    const v2f b2 = bp[4];  // k = 8+khalf, 9+khalf

    // ---- incremental 32-bit addressing state (one div pair at entry) ----
    const v2f* __restrict__ xv = (const v2f*)x;   // v2f-element view of x

    // A-operand row for this lane: r = wave*16 + icol (v2f index, all even)
    const int r0 = wave * 16 + icol;
    int av    = r0 % 25;
    int aoff2 = (base + (r0 / 25) * S_C + av * S_V + khalf) >> 1;
    // store base row for this lane: sr = wave*16 + mbase, column icol
    const int sr0 = wave * 16 + mbase;
    int sv    = sr0 % 25;
    int soff  = base + (sr0 / 25) * S_C + sv * S_V + icol;

    const bool do_store = (icol < 12);

    for (int tile = wave; tile < 100; tile += 8) {
        const v2f a0 = xv[aoff2];
        const v2f a1 = xv[aoff2 + 2];
        const v2f a2 = xv[aoff2 + 4];

        // advance A-operand index to next tile (r += 128 = 5*25 + 3)
        aoff2 += R_STEP_OFF / 2;
        av    += 3;
        if (av >= 25) { av -= 25; aoff2 += V_WRAP_OFF / 2; }
        __builtin_prefetch((const float*)(xv + aoff2), 0, 3);

        v8f acc = {};
        acc = __builtin_amdgcn_wmma_f32_16x16x4_f32(false, a0, false, b0,
                                                    (short)0, acc, false, false);
        acc = __builtin_amdgcn_wmma_f32_16x16x4_f32(false, a1, false, b1,
                                                    (short)0, acc, false, false);
        acc = __builtin_amdgcn_wmma_f32_16x16x4_f32(false, a2, false, b2,
                                                    (short)0, acc, false, false);

        // D layout: lane holds column icol; VGPR g holds row mbase+g.
        // Rows are consecutive -> at most one v-wrap among the 8 stores:
        // offset_g = soff + g*12 + (g >= w ? V_WRAP_OFF : 0), all independent.
        if (do_store) {
            const int w = 25 - sv;
            #pragma unroll
            for (int g = 0; g < 8; ++g) {
                const int o = soff +
                    ((g >= w) ? (V_WRAP_OFF + g * S_V) : (g * S_V));
                out[o] = acc[g];
            }
        }

        // advance store base to next tile
        soff += R_STEP_OFF;
        sv   += 3;
        if (sv >= 25) { sv -= 25; soff += V_WRAP_OFF; }
    }
}

extern "C" void kernel_launch(void* const* d_in, const int* in_sizes, int n_in,
                              void* d_out, int out_size, void* d_ws, size_t ws_size,
                              hipStream_t stream) {
    const float* x   = (const float*)d_in[0];
    const float* lam = (const float*)d_in[1];
    float*       out = (float*)d_out;
    float*       Aws = (float*)d_ws;   // 4096 * 256 floats = 4 MB

    tag_build_adj<<<4096, 32, 0, stream>>>(x, lam, Aws);
    tag_apply_wmma<<<4096, 256, 0, stream>>>(x, Aws, out);
}